// BiGraphConv_23124103921910
// MI455X (gfx1250) — compile-verified
//
#include <hip/hip_runtime.h>

typedef __attribute__((ext_vector_type(2))) float v2f;
typedef __attribute__((ext_vector_type(8))) float v8f;

// ---------------------------------------------------------------------------
// Y[r][n] = sum_k X[r][k] * W[n][k]  (+ bias[n])     (i.e. Y = X @ W^T + b)
// X: [nrows,64], W: [64,64] row-major, Y: [nrows,64]
// One wave -> 16x64 output tile via V_WMMA_F32_16X16X4_F32 (fp32 WMMA).
// Block = 256 threads = 8 waves = 128 rows. W staged in LDS (16 KB).
//
// Fragment layouts (ISA 7.12.2, wave32):
//   A 16x4 f32 : lane L holds A[L&15][2*(L>>4) + j] in VGPR j   (j=0,1)
//   B 4x16 f32 : lane L holds B[2*(L>>4) + j][L&15] in VGPR j
//   C/D 16x16  : VGPR r, lanes 0-15 -> M=r, N=lane; lanes 16-31 -> M=r+8
// ---------------------------------------------------------------------------
template <bool ADD_BIAS>
__global__ __launch_bounds__(256) void gemm_xwt(const float* __restrict__ X,
                                                const float* __restrict__ W,
                                                const float* __restrict__ bias,
                                                float* __restrict__ Y,
                                                int nrows)
{
    __shared__ float sW[64 * 64];
    // cooperative load of the 16 KB weight matrix into LDS
    for (int i = threadIdx.x * 4; i < 64 * 64; i += 256 * 4) {
        *(float4*)(&sW[i]) = *(const float4*)(&W[i]);
    }
    __syncthreads();

    const int wave = threadIdx.x >> 5;
    const int lane = threadIdx.x & 31;
    const int m    = lane & 15;   // row-within-tile (A) / col (B,C)
    const int half = lane >> 4;   // selects K pair
    const int row0 = (blockIdx.x * 8 + wave) * 16;
    if (row0 >= nrows) return;    // wave-uniform branch: EXEC stays all-1s

    v8f acc0 = {}, acc1 = {}, acc2 = {}, acc3 = {};

    const float* xrow = X + (size_t)(row0 + m) * 64;

#pragma unroll
    for (int k0 = 0; k0 < 64; k0 += 4) {
        const int kk = k0 + half * 2;
        v2f a  = *(const v2f*)(xrow + kk);
        v2f b0 = *(const v2f*)(&sW[( 0 + m) * 64 + kk]);
        v2f b1 = *(const v2f*)(&sW[(16 + m) * 64 + kk]);
        v2f b2 = *(const v2f*)(&sW[(32 + m) * 64 + kk]);
        v2f b3 = *(const v2f*)(&sW[(48 + m) * 64 + kk]);
        acc0 = __builtin_amdgcn_wmma_f32_16x16x4_f32(false, a, false, b0, (short)0, acc0, false, false);
        acc1 = __builtin_amdgcn_wmma_f32_16x16x4_f32(false, a, false, b1, (short)0, acc1, false, false);
        acc2 = __builtin_amdgcn_wmma_f32_16x16x4_f32(false, a, false, b2, (short)0, acc2, false, false);
        acc3 = __builtin_amdgcn_wmma_f32_16x16x4_f32(false, a, false, b3, (short)0, acc3, false, false);
    }

    float bs0 = 0.f, bs1 = 0.f, bs2 = 0.f, bs3 = 0.f;
    if (ADD_BIAS) {
        bs0 = bias[ 0 + m];
        bs1 = bias[16 + m];
        bs2 = bias[32 + m];
        bs3 = bias[48 + m];
    }

#pragma unroll
    for (int r = 0; r < 8; ++r) {
        const int mm   = r + half * 8;
        float*    yrow = Y + (size_t)(row0 + mm) * 64;
        yrow[ 0 + m] = acc0[r] + bs0;
        yrow[16 + m] = acc1[r] + bs1;
        yrow[32 + m] = acc2[r] + bs2;
        yrow[48 + m] = acc3[r] + bs3;
    }
}

// ---------------------------------------------------------------------------
// Hardware fp32 global atomic add (no return -> STOREcnt), device scope so
// RMW resolves at L2 (out fits in the 192 MB L2).
// ---------------------------------------------------------------------------
__device__ __forceinline__ void gatomic_fadd(float* p, float v)
{
    asm volatile("global_atomic_add_f32 %0, %1, off scope:SCOPE_DEV"
                 :
                 : "v"(p), "v"(v)
                 : "memory");
}

// ---------------------------------------------------------------------------
// Per-edge: out[dst[e]] += h[src[e]] * w[e]
// 16 threads per edge, float4 per thread -> coalesced 256 B gather per edge.
// ---------------------------------------------------------------------------
__global__ __launch_bounds__(256) void scatter_edges(const float* __restrict__ h,
                                                     const int* __restrict__ src,
                                                     const int* __restrict__ dst,
                                                     const float* __restrict__ ew,
                                                     float* __restrict__ out,
                                                     int nedges)
{
    const int tid = blockIdx.x * 256 + threadIdx.x;
    const int e   = tid >> 4;
    if (e >= nedges) return;
    const int c = (tid & 15) * 4;

    const int   s = src[e];
    const int   d = dst[e];
    const float w = ew[e];

    const float4 hv = *(const float4*)(h + (size_t)s * 64 + c);
    float*       o  = out + (size_t)d * 64 + c;

    gatomic_fadd(o + 0, hv.x * w);
    gatomic_fadd(o + 1, hv.y * w);
    gatomic_fadd(o + 2, hv.z * w);
    gatomic_fadd(o + 3, hv.w * w);
}

// ---------------------------------------------------------------------------
// Inputs (setup_inputs order):
//   0: x_src  [100000,64] f32      1: x_dst [100000,64] f32
//   2: edge_index_sd [2,E] int32   3: edge_weight [E] f32
//   4: W_nei [64,64] f32           5: W_self [64,64] f32
//   6: b_self [64] f32
// Output: [100000,64] f32.  Workspace: h_src = x_src @ W_nei^T (25.6 MB).
// ---------------------------------------------------------------------------
extern "C" void kernel_launch(void* const* d_in, const int* in_sizes, int n_in,
                              void* d_out, int out_size, void* d_ws, size_t ws_size,
                              hipStream_t stream)
{
    const float* x_src  = (const float*)d_in[0];
    const float* x_dst  = (const float*)d_in[1];
    const int*   eidx   = (const int*)d_in[2];
    const float* ew     = (const float*)d_in[3];
    const float* W_nei  = (const float*)d_in[4];
    const float* W_self = (const float*)d_in[5];
    const float* b_self = (const float*)d_in[6];

    float* out = (float*)d_out;
    float* h   = (float*)d_ws;   // [n_src, 64] scratch

    const int n_src = in_sizes[0] / 64;
    const int n_dst = in_sizes[1] / 64;
    const int E     = in_sizes[3];

    const int* src = eidx;       // row 0 of [2,E]
    const int* dst = eidx + E;   // row 1 of [2,E]

    const dim3 blk(256);

    // h = x_src @ W_nei^T
    gemm_xwt<false><<<dim3((n_src + 127) / 128), blk, 0, stream>>>(x_src, W_nei, nullptr, h, n_src);
    // out = x_dst @ W_self^T + b_self
    gemm_xwt<true><<<dim3((n_dst + 127) / 128), blk, 0, stream>>>(x_dst, W_self, b_self, out, n_dst);
    // out[dst] += h[src] * w   (device-scope hardware f32 atomics, L2-resident)
    const long long tot    = (long long)E * 16;
    const int       blocks = (int)((tot + 255) / 256);
    scatter_edges<<<blocks, blk, 0, stream>>>(h, src, dst, ew, out, E);
}